// fmmLayerExpTrainablePer_59906203845036
// MI455X (gfx1250) — compile-verified
//
#include <hip/hip_runtime.h>
#include <cstdint>

typedef float v2f __attribute__((ext_vector_type(2)));
typedef float v8f __attribute__((ext_vector_type(8)));

#define NTOT 2048          // 128 cells * 16 particles
#define LBOX 128.0f
#define INV_LBOX 0.0078125f   // 1/128 exact

// Block = 256 threads = 8 waves (wave32). Each block handles one s and 32 rows.
// Wave w: tile = w>>2 (16 rows each), j-chunk = w&3 (512 j's each).
// Per wave: WMMA f32 16x16x4 with B=ones reduces 16 rows x 4 j's per issue.
__global__ __launch_bounds__(256) void fmm_wmma_kernel(
    const float* __restrict__ Rin, const float* __restrict__ mu,
    const float* __restrict__ stdv, const float* __restrict__ biasv,
    float* __restrict__ out)
{
  __shared__ float Rsh[NTOT];
  __shared__ float partial[8][64];   // per wave: 16 rows x 4 kernels

  const int s    = blockIdx.x >> 6;  // 64 blocks per s
  const int blk  = blockIdx.x & 63;
  const int row0 = blk * 32;
  const float* __restrict__ R = Rin + s * NTOT;
  const int tid = threadIdx.x;

  // --- Stage R[s,:] (8 KB) into LDS via CDNA5 async copy (2 x B128 / thread) ---
  {
    uint32_t l0 = (uint32_t)(uintptr_t)(&Rsh[tid * 4]);
    uint64_t g0 = (uint64_t)(uintptr_t)(R + tid * 4);
    asm volatile("global_load_async_to_lds_b128 %0, %1, off"
                 :: "v"(l0), "v"(g0) : "memory");
    uint32_t l1 = (uint32_t)(uintptr_t)(&Rsh[tid * 4 + 1024]);
    uint64_t g1 = (uint64_t)(uintptr_t)(R + tid * 4 + 1024);
    asm volatile("global_load_async_to_lds_b128 %0, %1, off"
                 :: "v"(l1), "v"(g1) : "memory");
    asm volatile("s_wait_asynccnt 0" ::: "memory");
  }
  __syncthreads();

  const int wave  = tid >> 5;
  const int lane  = tid & 31;
  const int tile  = wave >> 2;        // 0..1
  const int chunk = wave & 3;         // 0..3  -> j in [chunk*512, chunk*512+512)
  const int m     = lane & 15;        // A-matrix row (M) for this lane
  const int iglob = row0 + tile * 16 + m;
  const float ri  = Rsh[iglob];
  const float mu0 = mu[0];
  // A layout (f32 16x4): lanes 0-15 hold K={0,1}, lanes 16-31 hold K={2,3}
  const int jcol = (lane >> 4) << 1;  // 0 or 2

  v8f acc0 = (v8f)0.0f, acc1 = (v8f)0.0f, acc2 = (v8f)0.0f, acc3 = (v8f)0.0f;
  const v2f bones = {1.0f, 1.0f};     // B = ones(4x16): layout-independent

  const int jend = chunk * 512 + 512;
  for (int jb = chunk * 512; jb < jend; jb += 4) {
    v2f a0, a1, a2, a3;
#pragma unroll
    for (int u = 0; u < 2; ++u) {
      const int j   = jb + jcol + u;
      const float a = Rsh[j] - ri;
      const float w = fmaf(-LBOX, rintf(a * INV_LBOX), a);  // periodic wrap (rndne)
      const bool diag = (j == iglob);
      const float msk = diag ? 0.0f : 1.0f;
      const float d   = diag ? 1.0f : fabsf(w);             // Dsafe trick
      const float inv = __builtin_amdgcn_rcpf(d);
      a0[u] = inv * msk;                                    // 1/r
      a1[u] = __builtin_amdgcn_rsqf(d) * msk;               // 1/sqrt(r)
      a2[u] = inv * inv * msk;                              // 1/r^2
      a3[u] = __expf(-mu0 * d) * msk;                       // exp(-mu r)
    }
    // D(16x16,f32) = A(16x4,f32) * ones(4x16) + C  => every column = row sum
    acc0 = __builtin_amdgcn_wmma_f32_16x16x4_f32(false, a0, false, bones,
                                                 (short)0, acc0, false, false);
    acc1 = __builtin_amdgcn_wmma_f32_16x16x4_f32(false, a1, false, bones,
                                                 (short)0, acc1, false, false);
    acc2 = __builtin_amdgcn_wmma_f32_16x16x4_f32(false, a2, false, bones,
                                                 (short)0, acc2, false, false);
    acc3 = __builtin_amdgcn_wmma_f32_16x16x4_f32(false, a3, false, bones,
                                                 (short)0, acc3, false, false);
  }

  // C/D layout: VGPR v -> row v (lanes 0-15) / row v+8 (lanes 16-31); all
  // lanes in a half hold the identical row sum (B was all ones).
  const int half = lane >> 4;
  if ((lane & 15) == 0) {             // lane 0 writes rows 0-7, lane 16 rows 8-15
#pragma unroll
    for (int v = 0; v < 8; ++v) {
      const int row = v + half * 8;
      partial[wave][row * 4 + 0] = acc0[v];
      partial[wave][row * 4 + 1] = acc1[v];
      partial[wave][row * 4 + 2] = acc2[v];
      partial[wave][row * 4 + 3] = acc3[v];
    }
  }
  __syncthreads();

  // Combine 4 j-chunks per tile, apply |std|*(sum - (N-1)*bias), write out.
  if (tid < 128) {
    const int t2 = tid >> 6;          // tile 0..1
    const int rk = tid & 63;          // row*4 + k
    const float ssum = partial[t2 * 4 + 0][rk] + partial[t2 * 4 + 1][rk] +
                       partial[t2 * 4 + 2][rk] + partial[t2 * 4 + 3][rk];
    const int row = rk >> 2;
    const int k   = rk & 3;
    const float val = fabsf(stdv[k]) * (ssum - 2047.0f * biasv[k]);
    const int i = row0 + t2 * 16 + row;
    out[(s * NTOT + i) * 4 + k] = val;
  }
}

extern "C" void kernel_launch(void* const* d_in, const int* in_sizes, int n_in,
                              void* d_out, int out_size, void* d_ws, size_t ws_size,
                              hipStream_t stream) {
  const float* Rin   = (const float*)d_in[0];
  const float* mu    = (const float*)d_in[1];
  const float* stdv  = (const float*)d_in[2];
  const float* biasv = (const float*)d_in[3];
  float* out = (float*)d_out;
  const int Ns = in_sizes[0] / NTOT;          // 4
  dim3 grid(Ns * 64), block(256);
  hipLaunchKernelGGL(fmm_wmma_kernel, grid, block, 0, stream,
                     Rin, mu, stdv, biasv, out);
}